// Quantize_35570919145657
// MI455X (gfx1250) — compile-verified
//
#include <hip/hip_runtime.h>
#include <hip/hip_bf16.h>

// Problem constants (S=1 collapses the block structure).
#define NROWS   131072            // 32*64*64 flattened spatial cells
#define KCODES  512               // codebook entries
#define DIMV    64                // code vector dim
#define TILES_PER_BLOCK 8         // 8 waves per 256-thread block, 1 row-tile each
#define BLOCK_THREADS   256
#define NBLOCKS (NROWS / (TILES_PER_BLOCK * 16))   // 1024

// d_out layout (floats, tuple concatenated in return order):
//   [0, 8388608)                out (= quantize, straight-through forward)
//   [8388608]                   diff
//   [8388609, 8388609+131072)   embed_ind (as float)
//   [8519681]                   perplexity
#define OUT_ELEMS (NROWS * DIMV)
#define DIFF_OFF  (OUT_ELEMS)
#define IND_OFF   (OUT_ELEMS + 1)
#define PERP_OFF  (OUT_ELEMS + 1 + NROWS)

typedef __attribute__((ext_vector_type(2))) float v2f;
typedef __attribute__((ext_vector_type(8))) float v8f;

// Main fused kernel: GEMM (fp32 WMMA) + argmin + gather + out + diff partials.
extern "C" __global__ __launch_bounds__(BLOCK_THREADS)
void vq_main(const float* __restrict__ x, const float* __restrict__ embed,
             float* __restrict__ outbuf, float* __restrict__ partial) {
  extern __shared__ float smem[];
  float* sEmb  = smem;                      // 32768 floats, layout [d*512 + k]
  float* sNorm = sEmb + DIMV * KCODES;      // 512 floats: 0.5*||e_k||^2
  float* sIdx  = sNorm + KCODES;            // 128 floats: per-wave winner rows
  float* sRed  = sIdx + TILES_PER_BLOCK*16; // 8 floats: per-wave diff sums

  const int tid = threadIdx.x;

  // Stage the whole codebook in LDS (128 KB of the 320 KB WGP LDS), coalesced.
  {
    const float4* g = (const float4*)embed;
    float4*       s = (float4*)sEmb;
    for (int i = tid; i < (DIMV * KCODES) / 4; i += BLOCK_THREADS) s[i] = g[i];
  }
  __syncthreads();

  // Half squared norms of each code column (argmin(dist) == argmax(f.e - 0.5||e||^2)).
  for (int k = tid; k < KCODES; k += BLOCK_THREADS) {
    float acc = 0.f;
    #pragma unroll
    for (int d = 0; d < DIMV; ++d) { float v = sEmb[d * KCODES + k]; acc += v * v; }
    sNorm[k] = 0.5f * acc;
  }
  __syncthreads();

  const int lane   = tid & 31;
  const int waveId = tid >> 5;
  const int half   = lane >> 4;   // 0: lanes 0-15, 1: lanes 16-31
  const int l15    = lane & 15;
  const int rowBase = (blockIdx.x * TILES_PER_BLOCK + waveId) * 16;

  // A fragments for V_WMMA_F32_16X16X4_F32:
  //   lane L<16: M=L, holds K={0,1}; lane L>=16: M=L-16, holds K={2,3}.
  // 16 chunks of K=4 cover the 64-dim contraction; kept in registers.
  v2f afrag[16];
  {
    const float* xrow = x + (size_t)(rowBase + l15) * DIMV + 2 * half;
    #pragma unroll
    for (int kk = 0; kk < 16; ++kk) {
      afrag[kk].x = xrow[4 * kk + 0];
      afrag[kk].y = xrow[4 * kk + 1];
    }
  }

  float bestVal[8];
  int   bestIdx[8];
  #pragma unroll
  for (int j = 0; j < 8; ++j) { bestVal[j] = -3.402823466e38f; bestIdx[j] = 0; }

  // Sweep 32 column tiles of 16 codes; fp32 WMMA accumulates 16x16 dots.
  for (int c = 0; c < KCODES / 16; ++c) {
    const int col = c * 16 + l15;           // absolute code index for this lane
    v8f acc = {};
    // B fragment (4x16): lane L: N=L&15, holds K={2*(L>>4), 2*(L>>4)+1}.
    const float* bbase = sEmb + (2 * half) * KCODES + col;
    #pragma unroll
    for (int kk = 0; kk < 16; ++kk) {
      v2f b;
      b.x = bbase[(4 * kk + 0) * KCODES];
      b.y = bbase[(4 * kk + 1) * KCODES];
      acc = __builtin_amdgcn_wmma_f32_16x16x4_f32(
          /*neg_a=*/false, afrag[kk], /*neg_b=*/false, b,
          /*c_mod=*/(short)0, acc, /*reuse_a=*/false, /*reuse_b=*/false);
    }
    const float adj = sNorm[col];
    // C layout: lanes 0-15 => N=lane, VGPR j => M=j; lanes 16-31 => M=j+8.
    #pragma unroll
    for (int j = 0; j < 8; ++j) {
      float s = acc[j] - adj;
      if (s > bestVal[j]) { bestVal[j] = s; bestIdx[j] = col; }  // ties keep lower idx
    }
  }

  // Cross-lane argmax within each 16-lane half (columns live across lanes).
  #pragma unroll
  for (int j = 0; j < 8; ++j) {
    float v = bestVal[j]; int i = bestIdx[j];
    #pragma unroll
    for (int m = 8; m >= 1; m >>= 1) {
      float ov = __shfl_xor(v, m, 32);
      int   oi = __shfl_xor(i, m, 32);
      if (ov > v || (ov == v && oi < i)) { v = ov; i = oi; }
    }
    bestVal[j] = v; bestIdx[j] = i;
  }

  // Lane 0 owns rows 0..7, lane 16 owns rows 8..15. Publish winners.
  if (l15 == 0) {
    #pragma unroll
    for (int j = 0; j < 8; ++j) {
      const int row = j + 8 * half;
      sIdx[waveId * 16 + row] = (float)bestIdx[j];
      outbuf[IND_OFF + rowBase + row] = (float)bestIdx[j];   // embed_ind as float
    }
  }
  // Same-wave DS store->load is in-order (DScnt); no block barrier needed here.

  // Gather codes from LDS, write out (coalesced float2), accumulate MSE.
  float dsum = 0.f;
  const int d0 = 2 * lane;   // lanes cover the 64 dims contiguously
  #pragma unroll
  for (int r = 0; r < 16; ++r) {
    const int idx = (int)sIdx[waveId * 16 + r];
    const float q0 = sEmb[(d0 + 0) * KCODES + idx];
    const float q1 = sEmb[(d0 + 1) * KCODES + idx];
    const float* xp = x + (size_t)(rowBase + r) * DIMV + d0;
    const float x0 = xp[0], x1 = xp[1];
    const float e0 = q0 - x0, e1 = q1 - x1;
    dsum += e0 * e0 + e1 * e1;
    float2 o; o.x = q0; o.y = q1;
    *(float2*)(outbuf + (size_t)(rowBase + r) * DIMV + d0) = o;
  }

  // Deterministic diff reduction: wave shuffle -> per-wave LDS -> per-block partial.
  #pragma unroll
  for (int m = 16; m >= 1; m >>= 1) dsum += __shfl_xor(dsum, m, 32);
  if (lane == 0) sRed[waveId] = dsum;
  __syncthreads();
  if (tid == 0) {
    float t = 0.f;
    #pragma unroll
    for (int w = 0; w < TILES_PER_BLOCK; ++w) t += sRed[w];
    partial[blockIdx.x] = t;
  }
}

// Fixed-order finalize: sum 1024 partials, write diff and the constant perplexity.
extern "C" __global__ __launch_bounds__(256)
void vq_finalize(const float* __restrict__ partial, float* __restrict__ outbuf) {
  __shared__ float s[256];
  const int t = threadIdx.x;
  float v = partial[t] + partial[t + 256] + partial[t + 512] + partial[t + 768];
  s[t] = v;
  __syncthreads();
  for (int stride = 128; stride >= 1; stride >>= 1) {
    if (t < stride) s[t] += s[t + stride];
    __syncthreads();
  }
  if (t == 0) {
    outbuf[DIFF_OFF] = s[0] / (float)OUT_ELEMS;
    // p = mean(one_hot) over ALL N*K entries is the scalar 1/K in the reference.
    const float p = 1.0f / (float)KCODES;
    outbuf[PERP_OFF] = __expf(-p * __logf(p + 1e-10f));
  }
}

extern "C" void kernel_launch(void* const* d_in, const int* in_sizes, int n_in,
                              void* d_out, int out_size, void* d_ws, size_t ws_size,
                              hipStream_t stream) {
  const float* x     = (const float*)d_in[0];   // [32,64,64,64] f32
  const float* embed = (const float*)d_in[1];   // [1,1,64,512]  f32
  float* outbuf  = (float*)d_out;
  float* partial = (float*)d_ws;                // 1024 floats of scratch

  const size_t shmem =
      (size_t)(DIMV * KCODES + KCODES + TILES_PER_BLOCK * 16 + TILES_PER_BLOCK) *
      sizeof(float);  // ~133.7 KB, well under the 320 KB WGP LDS

  (void)hipFuncSetAttribute((const void*)vq_main,
                            hipFuncAttributeMaxDynamicSharedMemorySize,
                            (int)shmem);

  vq_main<<<NBLOCKS, BLOCK_THREADS, shmem, stream>>>(x, embed, outbuf, partial);
  vq_finalize<<<1, 256, 0, stream>>>(partial, outbuf);
}